// GNN_RAG_Model_7189775254178
// MI455X (gfx1250) — compile-verified
//
#include <hip/hip_runtime.h>

// ---------------------------------------------------------------------------
// Problem constants (match reference)
// ---------------------------------------------------------------------------
#define NNODES 100000
#define NEDGES 1600000
#define DFEAT  128
#define NCLS   16
#define BN_EPS 1e-5f

typedef float v2f __attribute__((ext_vector_type(2)));
typedef float v8f __attribute__((ext_vector_type(8)));

// ---------------------------------------------------------------------------
// Degree count (float so we can use global_atomic_add_f32) and inversion
// ---------------------------------------------------------------------------
__global__ void count_deg_kernel(const int* __restrict__ dst,
                                 float* __restrict__ cnt, int ne) {
    int e = blockIdx.x * blockDim.x + threadIdx.x;
    if (e < ne) atomicAdd(&cnt[dst[e]], 1.0f);
}

__global__ void invert_cnt_kernel(float* __restrict__ cnt, int n) {
    int i = blockIdx.x * blockDim.x + threadIdx.x;
    if (i < n) cnt[i] = 1.0f / fmaxf(cnt[i], 1.0f);
}

// ---------------------------------------------------------------------------
// Fold BatchNorm(eval) + lin_l bias into per-channel affine:
//   BN(S + bl) = S*alpha + beta,  alpha = g*rsqrt(v+eps),
//   beta = (bl - m)*alpha + be
// ---------------------------------------------------------------------------
__global__ void prep_affine_kernel(const float* __restrict__ bl,
                                   const float* __restrict__ g,
                                   const float* __restrict__ be,
                                   const float* __restrict__ m,
                                   const float* __restrict__ v,
                                   float* __restrict__ alpha,
                                   float* __restrict__ beta) {
    int j = threadIdx.x;  // 128 threads
    float s  = g[j] * rsqrtf(v[j] + BN_EPS);
    alpha[j] = s;
    beta[j]  = (bl[j] - m[j]) * s + be[j];
}

// ---------------------------------------------------------------------------
// Edge scatter: one wave per edge; lane l moves features [4l, 4l+4) via
// float4 load + 4 global f32 atomics (L2 atomic units).
// ---------------------------------------------------------------------------
__global__ __launch_bounds__(256)
void scatter_add_kernel(const float* __restrict__ x,
                        const int* __restrict__ src,
                        const int* __restrict__ dst,
                        float* __restrict__ agg, int ne) {
    int edge = blockIdx.x * (blockDim.x >> 5) + (threadIdx.x >> 5);
    int lane = threadIdx.x & 31;
    if (edge >= ne) return;
    int s = src[edge];
    int d = dst[edge];
    float4 val = ((const float4*)(x + (size_t)s * DFEAT))[lane];
    float* o = agg + (size_t)d * DFEAT + lane * 4;
    atomicAdd(o + 0, val.x);
    atomicAdd(o + 1, val.y);
    atomicAdd(o + 2, val.z);
    atomicAdd(o + 3, val.w);
}

// ---------------------------------------------------------------------------
// Fused SAGE layer GEMM:
//   out = ReLU( (mean@wl + x@wr) * alpha + beta ),  mean = agg * inv_cnt
// Block: 256 thr = 8 waves; block handles 16 nodes x 128 outputs.
// Each wave: one 16x16 tile via V_WMMA_F32_16X16X4_F32, K-loop 32x2 WMMAs.
// A tiles staged in LDS (pitch 132 floats to avoid 64-bank conflicts).
// ---------------------------------------------------------------------------
#define APITCH 132

__global__ __launch_bounds__(256)
void sage_fused_kernel(const float* __restrict__ agg,
                       const float* __restrict__ x,
                       const float* __restrict__ inv,
                       const float* __restrict__ wl,
                       const float* __restrict__ wr,
                       const float* __restrict__ alpha,
                       const float* __restrict__ beta,
                       float* __restrict__ out) {
    __shared__ float As[16 * APITCH];
    __shared__ float Xs[16 * APITCH];

    const int tid    = threadIdx.x;
    const int blkrow = blockIdx.x * 16;

    // Cooperative stage of A (scaled mean) and X tiles: 16x128 each,
    // 512 float4 per matrix, 256 threads -> 2 float4 each.
    #pragma unroll
    for (int i = 0; i < 2; ++i) {
        int idx = tid * 2 + i;
        int r   = idx >> 5;        // row 0..15
        int c4  = idx & 31;        // float4 column 0..31
        float s = inv[blkrow + r];
        float4 av = ((const float4*)(agg + (size_t)(blkrow + r) * DFEAT))[c4];
        av.x *= s; av.y *= s; av.z *= s; av.w *= s;
        *(float4*)&As[r * APITCH + c4 * 4] = av;
        float4 xv = ((const float4*)(x + (size_t)(blkrow + r) * DFEAT))[c4];
        *(float4*)&Xs[r * APITCH + c4 * 4] = xv;
    }
    __syncthreads();

    const int wave  = tid >> 5;
    const int lane  = tid & 31;
    const int c0    = wave * 16;        // output-column tile base
    const int lrow  = lane & 15;        // row (A) / column (B,C) within tile
    const int khalf = (lane >> 4) * 2;  // lanes 16-31 carry K+2,K+3

    v8f acc = {};
    for (int k0 = 0; k0 < DFEAT; k0 += 4) {
        const int ka = k0 + khalf;
        v2f a, b;
        // mean @ wl
        a.x = As[lrow * APITCH + ka];
        a.y = As[lrow * APITCH + ka + 1];
        b.x = wl[(size_t)ka * DFEAT + c0 + lrow];
        b.y = wl[(size_t)(ka + 1) * DFEAT + c0 + lrow];
        acc = __builtin_amdgcn_wmma_f32_16x16x4_f32(false, a, false, b,
                                                    (short)0, acc, false, false);
        // x @ wr
        a.x = Xs[lrow * APITCH + ka];
        a.y = Xs[lrow * APITCH + ka + 1];
        b.x = wr[(size_t)ka * DFEAT + c0 + lrow];
        b.y = wr[(size_t)(ka + 1) * DFEAT + c0 + lrow];
        acc = __builtin_amdgcn_wmma_f32_16x16x4_f32(false, a, false, b,
                                                    (short)0, acc, false, false);
    }

    // Epilogue: fused affine(BN+bias) + ReLU.
    // C/D layout: VGPR r -> M=r (lanes 0-15) / M=r+8 (lanes 16-31), N=lane&15.
    const int col   = c0 + lrow;
    const float al  = alpha[col];
    const float bt  = beta[col];
    const int mbase = (lane >> 4) * 8;
    #pragma unroll
    for (int r = 0; r < 8; ++r) {
        float v = acc[r] * al + bt;
        v = fmaxf(v, 0.0f);
        out[(size_t)(blkrow + mbase + r) * DFEAT + col] = v;
    }
}

// ---------------------------------------------------------------------------
// Plain linear layer via WMMA: out = act(A @ W + bias)
// NCOLS = 128 (wf, 8 waves/block) or 16 (wc, 1 wave/block).
// ---------------------------------------------------------------------------
template <int NCOLS, bool RELU>
__global__ void linear_wmma_kernel(const float* __restrict__ A,
                                   const float* __restrict__ W,
                                   const float* __restrict__ bias,
                                   float* __restrict__ out) {
    const int tid    = threadIdx.x;
    const int wave   = tid >> 5;
    const int lane   = tid & 31;
    const int c0     = wave * 16;
    const int blkrow = blockIdx.x * 16;
    const int lrow   = lane & 15;
    const int khalf  = (lane >> 4) * 2;

    const float* Arow = A + (size_t)(blkrow + lrow) * DFEAT;

    v8f acc = {};
    for (int k0 = 0; k0 < DFEAT; k0 += 4) {
        const int ka = k0 + khalf;
        v2f a, b;
        a.x = Arow[ka];
        a.y = Arow[ka + 1];
        b.x = W[(size_t)ka * NCOLS + c0 + lrow];
        b.y = W[(size_t)(ka + 1) * NCOLS + c0 + lrow];
        acc = __builtin_amdgcn_wmma_f32_16x16x4_f32(false, a, false, b,
                                                    (short)0, acc, false, false);
    }

    const int col   = c0 + lrow;
    const float bt  = bias[col];
    const int mbase = (lane >> 4) * 8;
    #pragma unroll
    for (int r = 0; r < 8; ++r) {
        float v = acc[r] + bt;
        if (RELU) v = fmaxf(v, 0.0f);
        out[(size_t)(blkrow + mbase + r) * NCOLS + col] = v;
    }
}

// ---------------------------------------------------------------------------
// Host-side orchestration (graph-capture safe: async memsets + launches only)
// ---------------------------------------------------------------------------
extern "C" void kernel_launch(void* const* d_in, const int* in_sizes, int n_in,
                              void* d_out, int out_size, void* d_ws, size_t ws_size,
                              hipStream_t stream) {
    (void)in_sizes; (void)n_in; (void)out_size; (void)ws_size;

    const float* x  = (const float*)d_in[0];
    const int*   ei = (const int*)d_in[1];
    const int* src = ei;
    const int* dst = ei + NEDGES;

    // Per-layer params: base index 2 + 7*(layer): wl, bl, wr, g, be, m, v
    const float* wl_[3]; const float* bl_[3]; const float* wr_[3];
    const float* g_[3];  const float* be_[3]; const float* m_[3]; const float* v_[3];
    for (int l = 0; l < 3; ++l) {
        int b = 2 + 7 * l;
        wl_[l] = (const float*)d_in[b + 0];
        bl_[l] = (const float*)d_in[b + 1];
        wr_[l] = (const float*)d_in[b + 2];
        g_[l]  = (const float*)d_in[b + 3];
        be_[l] = (const float*)d_in[b + 4];
        m_[l]  = (const float*)d_in[b + 5];
        v_[l]  = (const float*)d_in[b + 6];
    }
    const float* wf = (const float*)d_in[23];
    const float* bf = (const float*)d_in[24];
    const float* wc = (const float*)d_in[25];
    const float* bc = (const float*)d_in[26];

    // Workspace carve-up (floats)
    float* ws  = (float*)d_ws;
    float* inv = ws;                                   // N
    float* agg = inv + NNODES;                         // N*128
    float* hA  = agg + (size_t)NNODES * DFEAT;         // N*128
    float* hB  = hA  + (size_t)NNODES * DFEAT;         // N*128
    float* aff = hB  + (size_t)NNODES * DFEAT;         // 6*128 (alpha/beta x3)

    const int ROWBLKS = NNODES / 16;  // 6250 exact

    // 1) Degrees (graph-invariant across the 3 layers) -> inverse counts
    hipMemsetAsync(inv, 0, NNODES * sizeof(float), stream);
    count_deg_kernel<<<(NEDGES + 255) / 256, 256, 0, stream>>>(dst, inv, NEDGES);
    invert_cnt_kernel<<<(NNODES + 255) / 256, 256, 0, stream>>>(inv, NNODES);

    // 2) Fold BN + lin_l bias into per-channel affine for each layer
    for (int l = 0; l < 3; ++l)
        prep_affine_kernel<<<1, 128, 0, stream>>>(bl_[l], g_[l], be_[l], m_[l], v_[l],
                                                  aff + 256 * l, aff + 256 * l + 128);

    // 3) Three fused SAGE layers
    const float* lin[3] = { x, hA, hB };
    float*       lout[3] = { hA, hB, hA };
    for (int l = 0; l < 3; ++l) {
        hipMemsetAsync(agg, 0, (size_t)NNODES * DFEAT * sizeof(float), stream);
        scatter_add_kernel<<<(NEDGES + 7) / 8, 256, 0, stream>>>(lin[l], src, dst,
                                                                 agg, NEDGES);
        sage_fused_kernel<<<ROWBLKS, 256, 0, stream>>>(agg, lin[l], inv,
                                                       wl_[l], wr_[l],
                                                       aff + 256 * l,
                                                       aff + 256 * l + 128,
                                                       lout[l]);
    }

    // 4) MLP head: ReLU(h @ wf + bf) -> logits = h @ wc + bc
    linear_wmma_kernel<DFEAT, true><<<ROWBLKS, 256, 0, stream>>>(hA, wf, bf, hB);
    linear_wmma_kernel<NCLS, false><<<ROWBLKS, 32, 0, stream>>>(hB, wc, bc,
                                                                (float*)d_out);
}